// ES2D_43516608643215
// MI455X (gfx1250) — compile-verified
//
#include <hip/hip_runtime.h>
#include <hip/hip_bf16.h>
#include <math.h>

typedef __attribute__((ext_vector_type(2))) float v2f;
typedef __attribute__((ext_vector_type(8))) float v8f;

#define B_  2
#define D_  96
#define H_  128
#define W_  128
#define K_  4
#define N_  8
#define R_  6
#define L_  4096   // (H/2)*(W/2)

// ---------------------------------------------------------------------------
// Kernel 1: efficient_scan gather  x(B,D,H,W) -> xs(B,K,D,L)
//   k0: l=i*64+j -> x[2i,2j]      k1: -> x[2j+1,2i]
//   k2: ->         x[2i,2j+1]     k3: -> x[2j+1,2i+1]
// ---------------------------------------------------------------------------
__global__ __launch_bounds__(256) void ss2d_gather(const float* __restrict__ x,
                                                   float* __restrict__ xs)
{
    int idx = blockIdx.x * 256 + threadIdx.x;          // over B*K*D*L (exact)
    int l = idx & (L_ - 1);
    int t = idx >> 12;                                 // (b*K+k)*D + d
    int d = t % D_;
    int bk = t / D_;
    int k = bk & 3;
    int b = bk >> 2;
    int i = l >> 6, j = l & 63;
    int h, w;
    if (k == 0)      { h = 2*i;     w = 2*j;     }
    else if (k == 1) { h = 2*j + 1; w = 2*i;     }
    else if (k == 2) { h = 2*i;     w = 2*j + 1; }
    else             { h = 2*j + 1; w = 2*i + 1; }
    xs[idx] = x[(((size_t)b * D_ + d) * H_ + h) * W_ + w];
}

// ---------------------------------------------------------------------------
// Kernel 2: WMMA projections. Per block: one (b,k) and a 64-column L-slab.
//   GEMM1: x_proj_weight[k] (22x96) @ xs_slab (96x64) -> x_dbl (22x64)
//          rows 0..5 -> dts_raw (LDS), 6..13 -> Bs, 14..21 -> Cs
//   GEMM2: dt_projs_weight[k] (96x6,pad8) @ dts_raw (8x64) -> dts (96x64)
// 4 waves, each owns 16 columns. Uses V_WMMA_F32_16X16X4_F32.
// ---------------------------------------------------------------------------
__global__ __launch_bounds__(128) void ss2d_proj_wmma(
    const float* __restrict__ xs,
    const float* __restrict__ xpw,    // (K,22,96)
    const float* __restrict__ dtw,    // (K,96,6)
    float* __restrict__ dts,          // (B,K,96,L)
    float* __restrict__ Bsg,          // (B,K,8,L)
    float* __restrict__ Csg)          // (B,K,8,L)
{
    __shared__ float xs_t[D_ * 64];     // 96 x 64 slab of xs
    __shared__ float w_t[32 * D_];      // x_proj weights, rows padded 22->32
    __shared__ float dtw_t[D_ * 8];     // dt weights, K padded 6->8
    __shared__ float raw_t[8 * 64];     // dts_raw, rows padded 6->8

    const int tid = threadIdx.x;
    const int k   = blockIdx.y;
    const int b   = blockIdx.z;
    const int l0  = blockIdx.x * 64;
    const size_t bk = (size_t)b * K_ + k;
    const size_t xsbase = bk * D_ * L_ + l0;

    for (int t = tid; t < D_ * 64; t += 128) {
        int d = t >> 6, c = t & 63;
        xs_t[t] = xs[xsbase + (size_t)d * L_ + c];
    }
    for (int t = tid; t < 32 * D_; t += 128) {
        int c = t / D_, d = t % D_;
        w_t[t] = (c < R_ + 2 * N_) ? xpw[((size_t)k * 22 + c) * D_ + d] : 0.f;
    }
    for (int t = tid; t < D_ * 8; t += 128) {
        int d = t >> 3, r = t & 7;
        dtw_t[t] = (r < R_) ? dtw[((size_t)k * D_ + d) * R_ + r] : 0.f;
    }
    for (int t = tid; t < 2 * 64; t += 128) raw_t[R_ * 64 + t - 0] = 0.f; // rows 6,7
    __syncthreads();

    const int lane = tid & 31;
    const int wave = tid >> 5;
    const int lm   = lane & 15;        // N / M position
    const int lh   = lane >> 4;        // half-wave K split
    const int col  = wave * 16 + lm;   // column in slab
    const size_t lglob = (size_t)l0 + col;

    // ---- GEMM1 ----
    v8f acc0 = {0.f,0.f,0.f,0.f,0.f,0.f,0.f,0.f};
    v8f acc1 = {0.f,0.f,0.f,0.f,0.f,0.f,0.f,0.f};
#pragma unroll
    for (int kk = 0; kk < D_; kk += 4) {
        v2f bf;
        bf.x = xs_t[(kk + 2*lh + 0) * 64 + col];
        bf.y = xs_t[(kk + 2*lh + 1) * 64 + col];
        v2f a0 = *(const v2f*)&w_t[( 0 + lm) * D_ + kk + 2*lh];
        v2f a1 = *(const v2f*)&w_t[(16 + lm) * D_ + kk + 2*lh];
        acc0 = __builtin_amdgcn_wmma_f32_16x16x4_f32(false, a0, false, bf,
                                                     (short)0, acc0, false, false);
        acc1 = __builtin_amdgcn_wmma_f32_16x16x4_f32(false, a1, false, bf,
                                                     (short)0, acc1, false, false);
    }
#pragma unroll
    for (int v = 0; v < 8; ++v) {
        int c = v + 8 * lh;                              // 0..15
        float val = acc0[v];
        if (c < R_)            raw_t[c * 64 + col] = val;
        else if (c < R_ + N_)  Bsg[(bk * N_ + (c - R_)) * L_ + lglob] = val;
        else                   Csg[(bk * N_ + (c - R_ - N_)) * L_ + lglob] = val;
        int c2 = 16 + v + 8 * lh;                        // 16..31, valid <22 (all Cs)
        if (c2 < R_ + 2 * N_)
            Csg[(bk * N_ + (c2 - R_ - N_)) * L_ + lglob] = acc1[v];
    }
    __syncthreads();

    // ---- GEMM2 ----
#pragma unroll
    for (int mt = 0; mt < 6; ++mt) {
        v8f acc = {0.f,0.f,0.f,0.f,0.f,0.f,0.f,0.f};
#pragma unroll
        for (int ks = 0; ks < 2; ++ks) {
            int kk = ks * 4;
            v2f af = *(const v2f*)&dtw_t[(mt * 16 + lm) * 8 + kk + 2*lh];
            v2f bf;
            bf.x = raw_t[(kk + 2*lh + 0) * 64 + col];
            bf.y = raw_t[(kk + 2*lh + 1) * 64 + col];
            acc = __builtin_amdgcn_wmma_f32_16x16x4_f32(false, af, false, bf,
                                                        (short)0, acc, false, false);
        }
#pragma unroll
        for (int v = 0; v < 8; ++v) {
            int d = mt * 16 + v + 8 * lh;
            dts[(bk * D_ + d) * L_ + lglob] = acc[v];
        }
    }
}

// ---------------------------------------------------------------------------
// Kernel 3: selective scan. One lane per (b,k,d,n): 6144 lanes = 192 waves.
// Serial over L=4096; reduce y over N=8 via shfl_xor within aligned groups.
// ---------------------------------------------------------------------------
__global__ __launch_bounds__(256) void ss2d_scan(
    const float* __restrict__ xs,  const float* __restrict__ dts,
    const float* __restrict__ Bsg, const float* __restrict__ Csg,
    const float* __restrict__ Alog, const float* __restrict__ Dsv,
    const float* __restrict__ dtb, float* __restrict__ y)
{
    int gid = blockIdx.x * 256 + threadIdx.x;      // (bk*D + d)*8 + n, exact grid
    int n  = gid & 7;
    int t  = gid >> 3;                             // bk*96 + d
    int d  = t % D_;
    int bk = t / D_;
    int kd = (bk & 3) * D_ + d;                    // row of A_logs/Ds/bias

    float Aval = -__expf(Alog[kd * N_ + n]);
    float bias = dtb[kd];
    float Dval = Dsv[kd];

    const size_t chan = (size_t)bk * D_ + d;
    const float* up = xs  + chan * L_;
    const float* dp = dts + chan * L_;
    const float* bp = Bsg + ((size_t)bk * N_ + n) * L_;
    const float* cp = Csg + ((size_t)bk * N_ + n) * L_;
    float* yp = y + chan * L_;

    float h = 0.f;
    for (int l = 0; l < L_; ++l) {
        if ((l & 63) == 0) {                       // global_prefetch_b8
            __builtin_prefetch(up + l + 64, 0, 0);
            __builtin_prefetch(dp + l + 64, 0, 0);
        }
        float u  = up[l];
        float dt = dp[l] + bias;
        float sp = (dt > 20.f) ? dt : __logf(1.f + __expf(dt));   // softplus
        float a  = __expf(sp * Aval);
        h = fmaf(a, h, sp * u * bp[l]);
        float yv = h * cp[l];
        yv += __shfl_xor(yv, 1, 32);
        yv += __shfl_xor(yv, 2, 32);
        yv += __shfl_xor(yv, 4, 32);
        if (n == 0) yp[l] = fmaf(Dval, u, yv);
    }
}

// ---------------------------------------------------------------------------
// Kernel 4: efficient_merge + LayerNorm. One wave per pixel (lanes over D=96).
// ---------------------------------------------------------------------------
__global__ __launch_bounds__(256) void ss2d_merge_ln(
    const float* __restrict__ y, const float* __restrict__ lnw,
    const float* __restrict__ lnb, float* __restrict__ out)
{
    int wid  = (blockIdx.x * 256 + threadIdx.x) >> 5;   // pixel id, exact grid
    int lane = threadIdx.x & 31;
    int w = wid & (W_ - 1);
    int h = (wid >> 7) & (H_ - 1);
    int b = wid >> 14;
    int k = (h & 1) + 2 * (w & 1);
    int l = (h & 1) ? ((w >> 1) * 64 + (h >> 1))
                    : ((h >> 1) * 64 + (w >> 1));

    const float* yp = y + ((size_t)b * K_ + k) * D_ * L_ + l;
    float v0 = yp[(size_t)(lane     ) * L_];
    float v1 = yp[(size_t)(lane + 32) * L_];
    float v2 = yp[(size_t)(lane + 64) * L_];

    float s = v0 + v1 + v2;
#pragma unroll
    for (int o = 16; o > 0; o >>= 1) s += __shfl_xor(s, o, 32);
    float mu = s * (1.f / 96.f);
    float d0 = v0 - mu, d1 = v1 - mu, d2 = v2 - mu;
    float q = d0*d0 + d1*d1 + d2*d2;
#pragma unroll
    for (int o = 16; o > 0; o >>= 1) q += __shfl_xor(q, o, 32);
    float inv = rsqrtf(q * (1.f / 96.f) + 1e-5f);

    size_t o0 = (size_t)wid * D_;
    out[o0 + lane     ] = d0 * inv * lnw[lane     ] + lnb[lane     ];
    out[o0 + lane + 32] = d1 * inv * lnw[lane + 32] + lnb[lane + 32];
    out[o0 + lane + 64] = d2 * inv * lnw[lane + 64] + lnb[lane + 64];
}

// ---------------------------------------------------------------------------
extern "C" void kernel_launch(void* const* d_in, const int* in_sizes, int n_in,
                              void* d_out, int out_size, void* d_ws, size_t ws_size,
                              hipStream_t stream)
{
    (void)in_sizes; (void)n_in; (void)out_size; (void)ws_size;
    const float* x    = (const float*)d_in[0];
    const float* xpw  = (const float*)d_in[1];
    const float* dtwp = (const float*)d_in[2];
    const float* dtb  = (const float*)d_in[3];
    const float* Alog = (const float*)d_in[4];
    const float* Dsv  = (const float*)d_in[5];
    const float* lnw  = (const float*)d_in[6];
    const float* lnb  = (const float*)d_in[7];
    // d_in[8] = step_size (==2, baked into the index math)

    float* ws  = (float*)d_ws;
    float* xs  = ws;                                   // B*K*D*L
    float* dts = xs  + (size_t)B_ * K_ * D_ * L_;      // B*K*D*L
    float* Bsg = dts + (size_t)B_ * K_ * D_ * L_;      // B*K*N*L
    float* Csg = Bsg + (size_t)B_ * K_ * N_ * L_;      // B*K*N*L
    float* yg  = Csg + (size_t)B_ * K_ * N_ * L_;      // B*K*D*L

    ss2d_gather<<<dim3((B_ * K_ * D_ * L_) / 256), dim3(256), 0, stream>>>(x, xs);
    ss2d_proj_wmma<<<dim3(L_ / 64, K_, B_), dim3(128), 0, stream>>>(
        xs, xpw, dtwp, dts, Bsg, Csg);
    ss2d_scan<<<dim3((B_ * K_ * D_ * N_) / 256), dim3(256), 0, stream>>>(
        xs, dts, Bsg, Csg, Alog, Dsv, dtb, yg);
    ss2d_merge_ln<<<dim3((B_ * H_ * W_ * 32) / 256), dim3(256), 0, stream>>>(
        yg, lnw, lnb, (float*)d_out);
}